// GNNStack_61100204753469
// MI455X (gfx1250) — compile-verified
//
#include <hip/hip_runtime.h>
#include <hip/hip_bf16.h>

// Problem constants (match reference)
#define NNODE  50000
#define NEDGE  800000
#define DIM    210      // D == C (H = 1); K of every GEMM in this net
#define KPAD   212      // DIM rounded up to a multiple of 4 (LDS zero-pad)
#define NLAYER 5
#define NOUT   128
#define NEG_SLOPE 0.2f

typedef float v2f __attribute__((ext_vector_type(2)));
typedef float v8f __attribute__((ext_vector_type(8)));

// ---------------------------------------------------------------------------
// Y[m, c] = sum_k A[m, k] * W[k, c] + bias[c]
// fp32 WMMA 16x16x4. One block = one 16-col tile of W (staged zero-padded in
// LDS, transposed) x 8 row tiles (one per wave). Requires n % 16 == 0 and
// K <= DIM (true for every GEMM here: K == 210).
// ---------------------------------------------------------------------------
__global__ void __launch_bounds__(256)
gemm_bias_wmma(const float* __restrict__ A, const float* __restrict__ W,
               const float* __restrict__ bias, float* __restrict__ Y,
               int n, int K, int Cc) {
  __shared__ float Bl[16][KPAD];          // Bl[n][k] = W[k, n0+n] (zero-padded)

  const int tid = threadIdx.x;
  const int n0  = blockIdx.x << 4;        // column-tile base

  // Stage + zero-pad the 16-column W tile, transposed so k is contiguous.
  for (int idx = tid; idx < 16 * KPAD; idx += 256) {
    const int nn  = idx / KPAD;
    const int kk  = idx - nn * KPAD;
    const int col = n0 + nn;
    float v = 0.f;
    if (kk < K && col < Cc) v = W[(size_t)kk * Cc + col];
    Bl[nn][kk] = v;
  }
  __syncthreads();

  const int lane = tid & 31;
  const int hi   = lane >> 4;             // selects K pair {0,1} vs {2,3}
  const int lo   = lane & 15;
  const int mt   = blockIdx.y * 8 + (tid >> 5);
  if (mt >= (n >> 4)) return;             // wave-uniform exit (no barriers after)

  const int rowA  = (mt << 4) + lo;       // A-fragment row (m)
  const int colB  = n0 + lo;              // B-fragment col (n)
  const float* __restrict__ arow = A + (size_t)rowA * (size_t)K;
  const float* __restrict__ brow = &Bl[lo][0];

  v8f acc = {0.f, 0.f, 0.f, 0.f, 0.f, 0.f, 0.f, 0.f};

  // Main loop: fully unguarded (k+3 < K), vector b64 loads from global + LDS.
  int k = 0;
  for (; k + 4 <= K; k += 4) {
    const int ka = k + (hi << 1);
    const v2f a = *(const v2f*)(arow + ka);   // a[v] = A[m][2*hi+v]
    const v2f b = *(const v2f*)(brow + ka);   // b[v] = W[2*hi+v][n]
    acc = __builtin_amdgcn_wmma_f32_16x16x4_f32(
        false, a, false, b, (short)0, acc, false, false);
  }
  // K tail (K % 4 != 0): clamp addresses, mask by multiply — branchless.
  if (k < K) {
    const int ka  = k + (hi << 1);
    const int ka0 = (ka     < K) ? ka     : K - 1;
    const int ka1 = (ka + 1 < K) ? ka + 1 : K - 1;
    v2f a, b;
    a.x = arow[ka0] * ((ka     < K) ? 1.f : 0.f);
    a.y = arow[ka1] * ((ka + 1 < K) ? 1.f : 0.f);
    b   = *(const v2f*)(brow + ka);           // LDS zero-padded to KPAD
    acc = __builtin_amdgcn_wmma_f32_16x16x4_f32(
        false, a, false, b, (short)0, acc, false, false);
  }

  if (colB < Cc) {
    const float bv = bias ? bias[colB] : 0.f;
    const int mbase = (mt << 4) + (hi << 3);
#pragma unroll
    for (int r = 0; r < 8; ++r)
      Y[(size_t)(mbase + r) * Cc + colB] = acc[r] + bv;  // D[r+8*hi][lo]
  }
}

// ---------------------------------------------------------------------------
// wrv[d] = sum_c Wr[d, c] * attr[c];  wrv[D] = sum_c br[c] * attr[c]
// ---------------------------------------------------------------------------
__global__ void fuse_attn_vec(const float* __restrict__ Wr,
                              const float* __restrict__ br,
                              const float* __restrict__ attr,
                              float* __restrict__ wrv, int D_, int C_) {
  const int d = blockIdx.x * blockDim.x + threadIdx.x;
  if (d < D_) {
    float s = 0.f;
    for (int c = 0; c < C_; ++c) s += Wr[(size_t)d * C_ + c] * attr[c];
    wrv[d] = s;
  }
  if (d == 0) {
    float s = 0.f;
    for (int c = 0; c < C_; ++c) s += br[c] * attr[c];
    wrv[D_] = s;
  }
}

// out[i] = cst + sum_c M[i, c] * v[c]
__global__ void rowdot(const float* __restrict__ M, const float* __restrict__ v,
                       const float* __restrict__ cstp, float* __restrict__ out,
                       int n, int C_) {
  const int i = blockIdx.x * blockDim.x + threadIdx.x;
  if (i >= n) return;
  float s = cstp ? cstp[0] : 0.f;
  const float* row = M + (size_t)i * C_;
  for (int c = 0; c < C_; ++c) s += row[c] * v[c];
  out[i] = s;
}

// Order-preserving float<->uint encoding for atomic segment-max.
__device__ __forceinline__ unsigned fenc(float f) {
  unsigned u = __float_as_uint(f);
  return (u & 0x80000000u) ? ~u : (u | 0x80000000u);
}
__device__ __forceinline__ float fdec(unsigned u) {
  return (u >= 0x80000000u) ? __uint_as_float(u & 0x7FFFFFFFu)
                            : __uint_as_float(~u);
}

__global__ void init_seg(unsigned* __restrict__ mEnc, int n) {
  const int i = blockIdx.x * blockDim.x + threadIdx.x;
  if (i < n) mEnc[i] = 0x007FFFFFu;   // fenc(-inf)
}

// aw[e] = leaky_relu(aL[src] + aR[dst]); segment max into mEnc[dst]
__global__ void edge_logits(const int* __restrict__ src, const int* __restrict__ dst,
                            const float* __restrict__ aL, const float* __restrict__ aR,
                            float* __restrict__ aw, unsigned* __restrict__ mEnc, int E_) {
  for (int i = blockIdx.x * blockDim.x + threadIdx.x; i < E_;
       i += gridDim.x * blockDim.x) {
    const int d = dst[i];
    float v = aL[src[i]] + aR[d];
    v = (v > 0.f) ? v : NEG_SLOPE * v;
    aw[i] = v;
    atomicMax(&mEnc[d], fenc(v));
  }
}

// ex[e] = exp(aw[e] - m[dst]);  denom[dst] += ex   (aw buffer reused as ex)
__global__ void edge_exp(const int* __restrict__ dst, float* __restrict__ awex,
                         const unsigned* __restrict__ mEnc,
                         float* __restrict__ denom, int E_) {
  for (int i = blockIdx.x * blockDim.x + threadIdx.x; i < E_;
       i += gridDim.x * blockDim.x) {
    const int d = dst[i];
    const float e = __expf(awex[i] - fdec(mEnc[d]));
    awex[i] = e;
    atomicAdd(&denom[d], e);
  }
}

// G[dst, :] += (ex / (denom[dst]+eps)) * S[src, :]   — one wave per edge
__global__ void edge_aggregate(const int* __restrict__ src, const int* __restrict__ dst,
                               const float* __restrict__ ex, const float* __restrict__ denom,
                               const float* __restrict__ S, float* __restrict__ G,
                               int E_, int C_) {
  const int lane = threadIdx.x & 31;
  const int epb = blockDim.x >> 5;                       // edges per block
  for (int e = blockIdx.x * epb + ((int)threadIdx.x >> 5); e < E_;
       e += gridDim.x * epb) {
    const int s = src[e], d = dst[e];
    const float alpha = ex[e] / (denom[d] + 1e-16f);
    const float* __restrict__ srow = S + (size_t)s * C_;
    float* __restrict__ grow = G + (size_t)d * C_;
    for (int c = lane; c < C_; c += 32)
      atomicAdd(&grow[c], alpha * srow[c]);
  }
}

__global__ void relu_copy(const float* __restrict__ in, float* __restrict__ out, size_t n) {
  const size_t i = (size_t)blockIdx.x * blockDim.x + threadIdx.x;
  if (i < n) out[i] = fmaxf(in[i], 0.f);
}

// ---------------------------------------------------------------------------
extern "C" void kernel_launch(void* const* d_in, const int* in_sizes, int n_in,
                              void* d_out, int out_size, void* d_ws, size_t ws_size,
                              hipStream_t stream) {
  const float* x_in  = (const float*)d_in[0];
  const int*   src   = (const int*)  d_in[1];
  const int*   dst   = (const int*)  d_in[2];
  const float* Wl    = (const float*)d_in[3];
  const float* bl    = (const float*)d_in[4];
  const float* Wr    = (const float*)d_in[5];
  const float* br    = (const float*)d_in[6];
  const float* att_l = (const float*)d_in[7];
  const float* att_r = (const float*)d_in[8];
  const float* W1    = (const float*)d_in[9];
  const float* b1    = (const float*)d_in[10];
  const float* W2    = (const float*)d_in[11];
  const float* b2    = (const float*)d_in[12];
  float* out = (float*)d_out;

  const size_t NC = (size_t)NNODE * DIM;

  // Workspace layout (fp32 slots)
  float*    X     = (float*)d_ws;          // [N, C]  layer input (layers >= 1)
  float*    S     = X + NC;                // [N, C]  source = x@Wl+bl
  float*    G     = S + NC;                // [N, C]  aggregated output
  float*    aL    = G + NC;                // [N]
  float*    aR    = aL + NNODE;            // [N]
  unsigned* mEnc  = (unsigned*)(aR + NNODE);           // [N]
  float*    denom = (float*)(mEnc + NNODE);            // [N]
  float*    awex  = denom + NNODE;         // [E] logits, then exp
  float*    wrv   = awex + NEDGE;          // [D+1] fused Wr@att_r (+ br·att_r)

  const int TPB = 256;

  auto launch_gemm = [&](const float* A, const float* W, const float* bias,
                         float* Y, int n, int K, int Cc) {
    dim3 grid((Cc + 15) / 16, ((n / 16) + 7) / 8);
    gemm_bias_wmma<<<grid, TPB, 0, stream>>>(A, W, bias, Y, n, K, Cc);
  };

  const int nodeBlocks = (NNODE + TPB - 1) / TPB;
  const int edgeBlocks = (NEDGE + TPB - 1) / TPB;
  const int epb = TPB / 32;
  const int aggBlocks = (NEDGE + epb - 1) / epb;

  const float* xin = x_in;
  for (int l = 0; l < NLAYER; ++l) {
    const float* Wl_l = Wl + (size_t)l * DIM * DIM;
    const float* bl_l = bl + (size_t)l * DIM;
    const float* Wr_l = Wr + (size_t)l * DIM * DIM;
    const float* br_l = br + (size_t)l * DIM;
    const float* atl  = att_l + (size_t)l * DIM;   // H == 1
    const float* atr  = att_r + (size_t)l * DIM;

    // source = x @ Wl + bl        (the only real GEMM of the layer)
    launch_gemm(xin, Wl_l, bl_l, S, NNODE, DIM, DIM);
    // wrv = Wr @ att_r ; wrv[D] = br . att_r   (collapses the lin_r GEMM)
    fuse_attn_vec<<<1, TPB, 0, stream>>>(Wr_l, br_l, atr, wrv, DIM, DIM);
    // alpha_l = source . att_l ; alpha_r = x . wrv + wrv[D]
    rowdot<<<nodeBlocks, TPB, 0, stream>>>(S, atl, nullptr, aL, NNODE, DIM);
    rowdot<<<nodeBlocks, TPB, 0, stream>>>(xin, wrv, wrv + DIM, aR, NNODE, DIM);

    // segment-softmax state
    hipMemsetAsync(G, 0, NC * sizeof(float), stream);
    hipMemsetAsync(denom, 0, NNODE * sizeof(float), stream);
    init_seg<<<nodeBlocks, TPB, 0, stream>>>(mEnc, NNODE);

    edge_logits<<<edgeBlocks, TPB, 0, stream>>>(src, dst, aL, aR, awex, mEnc, NEDGE);
    edge_exp<<<edgeBlocks, TPB, 0, stream>>>(dst, awex, mEnc, denom, NEDGE);
    edge_aggregate<<<aggBlocks, TPB, 0, stream>>>(src, dst, awex, denom, S, G,
                                                  NEDGE, DIM);

    // x_{l+1} = relu(G)
    const size_t ncBlocks = (NC + TPB - 1) / TPB;
    relu_copy<<<(int)ncBlocks, TPB, 0, stream>>>(G, X, NC);
    xin = X;
  }

  // post_mp: h = x @ W1 + b1 ; out = h @ W2 + b2   (no activation in between)
  launch_gemm(xin, W1, b1, S, NNODE, DIM, DIM);
  launch_gemm(S, W2, b2, out, NNODE, DIM, NOUT);
}